// baseline_Layers_34479997452525
// MI455X (gfx1250) — compile-verified
//
#include <hip/hip_runtime.h>

// Problem constants (from reference)
#define N_NODES 50000
#define N_REL   16
#define HID     64
#define N_LAB   32
#define N_EDGE  1600000
#define K_AGG   1024          // R*H
#define K_TOT   1088          // R*H + H (x@root2 folded into GEMM)

typedef float v2f __attribute__((ext_vector_type(2)));
typedef float v8f __attribute__((ext_vector_type(8)));

// Workspace layout (floats):
//   cnt  : [R*N]            per-(rel,dst) edge counts
//   Bmat : [K_TOT * L]      concat(weight2 flat, root2 flat)
//   lhs  : [N * K_TOT]      cols 0..1023 = agg2 (scaled sums), cols 1024..1087 = x
#define WS_CNT_OFF  0
#define WS_B_OFF    (N_REL * N_NODES)                 // 800000
#define WS_LHS_OFF  (WS_B_OFF + K_TOT * N_LAB)        // 834816 (even -> 8B aligned)
#define WS_TOTAL    (WS_LHS_OFF + (size_t)N_NODES * K_TOT)

__device__ __forceinline__ void fatomic_add(float* p, float v) {
  __hip_atomic_fetch_add(p, v, __ATOMIC_RELAXED, __HIP_MEMORY_SCOPE_AGENT);
}

__global__ void k_zero(float* __restrict__ p, unsigned long long n) {
  unsigned long long i = (unsigned long long)blockIdx.x * blockDim.x + threadIdx.x;
  unsigned long long stride = (unsigned long long)gridDim.x * blockDim.x;
  for (; i < n; i += stride) p[i] = 0.0f;
}

// Per-(rel,dst) segment counts
__global__ void k_count(const int* __restrict__ etype, const int* __restrict__ dst,
                        float* __restrict__ cnt) {
  unsigned e = blockIdx.x * blockDim.x + threadIdx.x;
  if (e >= N_EDGE) return;
  fatomic_add(&cnt[etype[e] * N_NODES + dst[e]], 1.0f);
}

// Bmat = concat(weight2[R,H,L] flat, root2[H,L] flat) -> [K_TOT, L] row-major
__global__ void k_bmat(const float* __restrict__ w2, const float* __restrict__ r2,
                       float* __restrict__ B) {
  unsigned t = blockIdx.x * blockDim.x + threadIdx.x;
  if (t >= K_TOT * N_LAB) return;
  B[t] = (t < K_AGG * N_LAB) ? w2[t] : r2[t - K_AGG * N_LAB];
}

// Layer 1: one wave32 per edge; lane handles 2 of H=64 features.
// agg1[dst][h] += weight1[rel,src][h] / cnt[rel,dst]   (mean-then-sum-over-rel fused)
// accumulated directly into lhs columns 1024..1087.
__global__ void k_layer1(const int* __restrict__ src_a, const int* __restrict__ dst_a,
                         const int* __restrict__ etype, const float* __restrict__ w1,
                         const float* __restrict__ cnt, float* __restrict__ lhs) {
  unsigned tid = blockIdx.x * blockDim.x + threadIdx.x;
  unsigned e = tid >> 5;
  unsigned lane = tid & 31;
  if (e >= N_EDGE) return;
  int rel = etype[e], s = src_a[e], d = dst_a[e];
  float inv = 1.0f / fmaxf(cnt[rel * N_NODES + d], 1.0f);
  const float* wrow = w1 + ((size_t)rel * N_NODES + (size_t)s) * HID + 2u * lane;
  v2f v = *(const v2f*)wrow;                           // coalesced 256B/wave
  float* o = lhs + (size_t)d * K_TOT + K_AGG + 2u * lane;
  fatomic_add(o + 0, v.x * inv);
  fatomic_add(o + 1, v.y * inv);
}

// x = relu(agg1 + root1 + bias1), in place in lhs cols 1024..1087
__global__ void k_relu(const float* __restrict__ root1, const float* __restrict__ bias1,
                       float* __restrict__ lhs) {
  unsigned idx = blockIdx.x * blockDim.x + threadIdx.x;
  if (idx >= N_NODES * HID) return;
  unsigned n = idx >> 6, h = idx & 63;
  size_t p = (size_t)n * K_TOT + K_AGG + h;
  lhs[p] = fmaxf(lhs[p] + root1[idx] + bias1[h], 0.0f);
}

// Layer 2: one wave32 per edge. agg2[dst][rel*H + h] += x[src][h] / cnt[rel,dst]
// x read from lhs cols 1024+ (12.8MB, L2-resident); writes to lhs cols < 1024.
__global__ void k_layer2(const int* __restrict__ src_a, const int* __restrict__ dst_a,
                         const int* __restrict__ etype, const float* __restrict__ cnt,
                         float* __restrict__ lhs) {
  unsigned tid = blockIdx.x * blockDim.x + threadIdx.x;
  unsigned e = tid >> 5;
  unsigned lane = tid & 31;
  if (e >= N_EDGE) return;
  int rel = etype[e], s = src_a[e], d = dst_a[e];
  float inv = 1.0f / fmaxf(cnt[rel * N_NODES + d], 1.0f);
  const float* xrow = lhs + (size_t)s * K_TOT + K_AGG + 2u * lane;
  v2f v = *(const v2f*)xrow;
  float* o = lhs + (size_t)d * K_TOT + (unsigned)rel * HID + 2u * lane;
  fatomic_add(o + 0, v.x * inv);
  fatomic_add(o + 1, v.y * inv);
}

// GEMM: out[N,32] = sigmoid( lhs[N,1088] @ Bmat[1088,32] + bias2 )
// One wave per 16-row tile; computes both 16-col output tiles, reusing the A frag.
// V_WMMA_F32_16X16X4_F32 layouts:
//   A (16x4):  lane<16: {A[m][k0],A[m][k0+1]}, lane>=16: {A[m][k0+2],A[m][k0+3]}, m=lane&15
//   B (4x16):  lane<16: {B[k0][n],B[k0+1][n]}, lane>=16: {B[k0+2][n],B[k0+3][n]}, n=lane&15
//   C/D 16x16: vgpr v, lane l -> row = v + 8*(l>>4), col = l&15
__global__ void k_gemm(const float* __restrict__ lhs, const float* __restrict__ B,
                       const float* __restrict__ bias2, float* __restrict__ out) {
  unsigned tid = blockIdx.x * blockDim.x + threadIdx.x;
  unsigned wave = tid >> 5;
  unsigned lane = tid & 31;
  if (wave >= N_NODES / 16) return;     // 3125 row tiles, uniform per wave (EXEC all-1)
  unsigned half = lane >> 4;            // 0 or 1
  unsigned sub = lane & 15;
  const float* arow = lhs + (size_t)(wave * 16u + sub) * K_TOT + 2u * half;
  v8f c0 = {}; v8f c1 = {};
  for (int k0 = 0; k0 < K_TOT; k0 += 4) {
    v2f a;  a.x = arow[k0];  a.y = arow[k0 + 1];
    int kb = k0 + 2 * (int)half;
    v2f b0; b0.x = B[kb * N_LAB + sub];        b0.y = B[(kb + 1) * N_LAB + sub];
    v2f b1; b1.x = B[kb * N_LAB + 16 + sub];   b1.y = B[(kb + 1) * N_LAB + 16 + sub];
    c0 = __builtin_amdgcn_wmma_f32_16x16x4_f32(false, a, false, b0, (short)0, c0, false, false);
    c1 = __builtin_amdgcn_wmma_f32_16x16x4_f32(false, a, false, b1, (short)0, c1, false, false);
  }
  #pragma unroll
  for (int v = 0; v < 8; v++) {
    unsigned row = wave * 16u + (unsigned)v + 8u * half;
    float x0 = c0[v] + bias2[sub];
    float x1 = c1[v] + bias2[16 + sub];
    out[row * N_LAB + sub]      = 1.0f / (1.0f + __expf(-x0));
    out[row * N_LAB + 16 + sub] = 1.0f / (1.0f + __expf(-x1));
  }
}

extern "C" void kernel_launch(void* const* d_in, const int* in_sizes, int n_in,
                              void* d_out, int out_size, void* d_ws, size_t ws_size,
                              hipStream_t stream) {
  const int* ei     = (const int*)d_in[0];       // [2, E]
  const int* src_a  = ei;
  const int* dst_a  = ei + N_EDGE;
  const int* etype  = (const int*)d_in[1];       // [E]
  const float* w1   = (const float*)d_in[2];     // [R, N, H]
  const float* r1   = (const float*)d_in[3];     // [N, H]
  const float* b1   = (const float*)d_in[4];     // [H]
  const float* w2   = (const float*)d_in[5];     // [R, H, L]
  const float* r2   = (const float*)d_in[6];     // [H, L]
  const float* b2   = (const float*)d_in[7];     // [L]
  float* out = (float*)d_out;

  float* ws   = (float*)d_ws;
  float* cnt  = ws + WS_CNT_OFF;
  float* Bmat = ws + WS_B_OFF;
  float* lhs  = ws + WS_LHS_OFF;

  (void)in_sizes; (void)n_in; (void)out_size; (void)ws_size;

  // 1) zero cnt + Bmat + lhs region (Bmat overwritten anyway)
  k_zero<<<2048, 256, 0, stream>>>(ws, (unsigned long long)WS_TOTAL);
  // 2) per-(rel,dst) counts
  k_count<<<(N_EDGE + 255) / 256, 256, 0, stream>>>(etype, dst_a, cnt);
  // 3) build combined B matrix [1088, 32]
  k_bmat<<<(K_TOT * N_LAB + 255) / 256, 256, 0, stream>>>(w2, r2, Bmat);
  // 4) layer-1 scatter (1 wave per edge)
  k_layer1<<<(N_EDGE * 32u + 255) / 256, 256, 0, stream>>>(src_a, dst_a, etype, w1, cnt, lhs);
  // 5) relu(agg1 + root1 + bias1) -> x stored in lhs cols 1024..1087
  k_relu<<<(N_NODES * HID + 255) / 256, 256, 0, stream>>>(r1, b1, lhs);
  // 6) layer-2 scatter (1 wave per edge)
  k_layer2<<<(N_EDGE * 32u + 255) / 256, 256, 0, stream>>>(src_a, dst_a, etype, cnt, lhs);
  // 7) WMMA GEMM + bias + sigmoid
  k_gemm<<<((N_NODES / 16) * 32 + 255) / 256, 256, 0, stream>>>(lhs, Bmat, b2, out);
}